// rsn_cluster_70437463654561
// MI455X (gfx1250) — compile-verified
//
#include <hip/hip_runtime.h>
#include <hip/hip_bf16.h>

typedef float v2f __attribute__((ext_vector_type(2)));
typedef float v8f __attribute__((ext_vector_type(8)));

#define B_   4
#define C_   16
#define H_   240
#define W_   320
#define P_   (H_*W_)
#define K_   100
#define KP_  112            // K padded to 7 tiles of 16
#define NIT_ 10
#define BW2_ (0.16f*0.16f)
#define PADV_ 1.0e3f        // pad-mean value: d2 ~ 1.6e7 >> bw2, no overflow/NaN

// workspace layout (float offsets)
#define OFF_MEANS 0
#define OFF_NUM   (B_*KP_*C_)
#define OFF_DEN   (OFF_NUM + B_*KP_*C_)
#define OFF_FN2   (OFF_DEN + B_*KP_)

// ---------------------------------------------------------------- init means
__global__ void ms_init(const float* __restrict__ feat, const int* __restrict__ seed,
                        float* __restrict__ ws) {
  int t = blockIdx.x * blockDim.x + threadIdx.x;
  if (t >= B_*KP_) return;
  int b = t / KP_, k = t % KP_;
  float* means = ws + OFF_MEANS;
  float* num   = ws + OFF_NUM;
  float* den   = ws + OFF_DEN;
  if (k < K_) {
    int idx = seed[b*K_ + k];
    #pragma unroll
    for (int c = 0; c < C_; ++c)
      means[t*C_ + c] = feat[(b*C_ + c)*P_ + idx];
  } else {
    #pragma unroll
    for (int c = 0; c < C_; ++c) means[t*C_ + c] = PADV_;
  }
  #pragma unroll
  for (int c = 0; c < C_; ++c) num[t*C_ + c] = 0.0f;
  den[t] = 0.0f;
}

// ---------------------------------------------------------------- fn2 = ||feat||^2 per pixel
__global__ void ms_fn2(const float* __restrict__ feat, float* __restrict__ ws) {
  int t = blockIdx.x * blockDim.x + threadIdx.x;  // t in [0, B*P)
  int b = t / P_, p = t % P_;
  float s = 0.0f;
  #pragma unroll
  for (int c = 0; c < C_; ++c) {
    float v = feat[(b*C_ + c)*P_ + p];
    s = fmaf(v, v, s);
  }
  ws[OFF_FN2 + t] = s;
}

// ---------------------------------------------------------------- WMMA accumulation pass
// grid (30, 7, B), block 32 (one wave). Each wave: one (b, k-tile), 160 p-tiles.
__global__ __launch_bounds__(32)
void ms_accum(const float* __restrict__ feat, float* __restrict__ ws) {
  const int lane = threadIdx.x;
  const int half = lane >> 4;          // 0: lanes 0-15, 1: lanes 16-31
  const int lm   = lane & 15;
  const int b  = blockIdx.z;
  const int k0 = blockIdx.y * 16;

  const float* means = ws + OFF_MEANS + (b*KP_ + k0)*C_;
  float*       num   = ws + OFF_NUM;
  float*       den   = ws + OFF_DEN;
  const float* fn2   = ws + OFF_FN2 + b*P_;
  const float* fb    = feat + b*C_*P_;

  __shared__ float smask[16*16];

  // A fragments: means tile (M=k rows on lanes, K=c split across V0/V1 x halves)
  v2f afrag[4];
  float mn2p = 0.0f;
  #pragma unroll
  for (int cb = 0; cb < 4; ++cb) {
    int c = 4*cb + 2*half;
    float x = means[lm*C_ + c];
    float y = means[lm*C_ + c + 1];
    afrag[cb].x = x; afrag[cb].y = y;
    mn2p = fmaf(x, x, fmaf(y, y, mn2p));
  }
  float mn2full = mn2p + __shfl_xor(mn2p, 16);        // lane j: ||mean_{j%16}||^2
  float mn2row[8];
  #pragma unroll
  for (int v = 0; v < 8; ++v)                          // D-row m = v + 8*half
    mn2row[v] = __shfl(mn2full, v + 8*half);

  v8f numAcc = {};
  float denAcc[8];
  #pragma unroll
  for (int v = 0; v < 8; ++v) denAcc[v] = 0.0f;

  const int tiles = P_ / (16 * 30);                    // 160
  const int t0 = blockIdx.x * tiles;

  for (int t = 0; t < tiles; ++t) {
    const int p0 = (t0 + t) * 16;

    // B fragments for GEMM1 (contraction c, N = p)
    v2f bfrag[4];
    #pragma unroll
    for (int cb = 0; cb < 4; ++cb) {
      int c = 4*cb + 2*half;
      bfrag[cb].x = fb[c*P_ + p0 + lm];
      bfrag[cb].y = fb[(c+1)*P_ + p0 + lm];
    }
    // B fragments for GEMM2 (contraction p, N = c): contiguous p pair -> b64
    v2f b2frag[4];
    #pragma unroll
    for (int pb = 0; pb < 4; ++pb)
      b2frag[pb] = *(const v2f*)(fb + lm*P_ + p0 + 4*pb + 2*half);

    float fn2v = fn2[p0 + lm];

    // GEMM1: dot[k,p] over C=16 -> 4 chained fp32 WMMAs
    v8f dot = {};
    #pragma unroll
    for (int cb = 0; cb < 4; ++cb)
      dot = __builtin_amdgcn_wmma_f32_16x16x4_f32(
          false, afrag[cb], false, bfrag[cb], (short)0, dot, false, false);

    // mask in D layout: lane -> (m = v+8*half, n = lm)
    #pragma unroll
    for (int v = 0; v < 8; ++v) {
      float d2 = fn2v + mn2row[v] - 2.0f * dot[v];
      float m  = (d2 < BW2_) ? 1.0f : 0.0f;
      denAcc[v] += m;
      smask[(v + 8*half)*16 + lm] = m;
    }
    __syncthreads();   // single-wave WG: lowers to dscnt wait + s_nop barrier

    // GEMM2: num[k,c] += mask(16x4 over p) x feat(4x16 over c)
    #pragma unroll
    for (int pb = 0; pb < 4; ++pb) {
      v2f a2 = *(const v2f*)(&smask[lm*16 + 4*pb + 2*half]);
      numAcc = __builtin_amdgcn_wmma_f32_16x16x4_f32(
          false, a2, false, b2frag[pb], (short)0, numAcc, false, false);
    }
    __syncthreads();   // protect smask WAR for next tile
  }

  // den: reduce over the 16 lanes of each half (rows m = v + 8*half)
  #pragma unroll
  for (int v = 0; v < 8; ++v) {
    float d = denAcc[v];
    d += __shfl_xor(d, 1); d += __shfl_xor(d, 2);
    d += __shfl_xor(d, 4); d += __shfl_xor(d, 8);
    if (lm == 0) atomicAdd(&den[b*KP_ + k0 + v + 8*half], d);
  }
  // num: D layout lane -> (k = v+8*half, c = lm)
  #pragma unroll
  for (int v = 0; v < 8; ++v)
    atomicAdd(&num[(b*KP_ + k0 + v + 8*half)*C_ + lm], numAcc[v]);
}

// ---------------------------------------------------------------- means update (+ re-zero acc)
__global__ void ms_update(float* __restrict__ ws) {
  int t = blockIdx.x * blockDim.x + threadIdx.x;
  if (t >= B_*KP_) return;
  float* means = ws + OFF_MEANS;
  float* num   = ws + OFF_NUM;
  float* den   = ws + OFF_DEN;
  float d = den[t];
  float inv = 1.0f / fmaxf(d, 1.0f);
  #pragma unroll
  for (int c = 0; c < C_; ++c) {
    float nv = num[t*C_ + c];
    if (d > 0.0f) means[t*C_ + c] = nv * inv;
    num[t*C_ + c] = 0.0f;
  }
  den[t] = 0.0f;
}

// ---------------------------------------------------------------- label assignment
__global__ __launch_bounds__(256)
void ms_labels(const float* __restrict__ feat, const float* __restrict__ ws,
               float* __restrict__ out) {
  __shared__ float sm[K_*C_];
  const int b = blockIdx.y;
  const float* means = ws + OFF_MEANS + b*KP_*C_;   // first K_ rows are real
  for (int i = threadIdx.x; i < K_*C_; i += blockDim.x) sm[i] = means[i];
  __syncthreads();

  const int p = blockIdx.x * blockDim.x + threadIdx.x;   // P_ exactly covered
  float f[C_];
  #pragma unroll
  for (int c = 0; c < C_; ++c) f[c] = feat[(b*C_ + c)*P_ + p];

  float bestd = 3.4e38f;
  int best = 0;
  for (int k = 0; k < K_; ++k) {
    float d = 0.0f;
    #pragma unroll
    for (int c = 0; c < C_; ++c) {
      float t = f[c] - sm[k*C_ + c];
      d = fmaf(t, t, d);
    }
    if (d < bestd) { bestd = d; best = k; }
  }
  out[b*P_ + p] = (bestd < BW2_) ? (float)(best + 1) : 0.0f;
}

// ---------------------------------------------------------------- emit final means
__global__ void ms_copy_means(const float* __restrict__ ws, float* __restrict__ out) {
  int t = blockIdx.x * blockDim.x + threadIdx.x;
  if (t >= B_*K_*C_) return;
  int b = t / (K_*C_);
  int r = t % (K_*C_);
  out[B_*P_ + t] = ws[OFF_MEANS + b*KP_*C_ + r];
}

// ---------------------------------------------------------------- launcher
extern "C" void kernel_launch(void* const* d_in, const int* in_sizes, int n_in,
                              void* d_out, int out_size, void* d_ws, size_t ws_size,
                              hipStream_t stream) {
  (void)in_sizes; (void)n_in; (void)out_size; (void)ws_size;
  const float* feat = (const float*)d_in[0];
  const int*   seed = (const int*)d_in[1];
  float* out = (float*)d_out;
  float* ws  = (float*)d_ws;

  ms_init<<<(B_*KP_ + 63)/64, 64, 0, stream>>>(feat, seed, ws);
  ms_fn2<<<(B_*P_)/256, 256, 0, stream>>>(feat, ws);

  for (int it = 0; it < NIT_; ++it) {
    ms_accum<<<dim3(30, KP_/16, B_), 32, 0, stream>>>(feat, ws);
    ms_update<<<(B_*KP_ + 63)/64, 64, 0, stream>>>(ws);
  }

  ms_labels<<<dim3(P_/256, B_), 256, 0, stream>>>(feat, ws, out);
  ms_copy_means<<<(B_*K_*C_ + 255)/256, 256, 0, stream>>>(ws, out);
}